// SimulatedRingAttention_51531017617695
// MI455X (gfx1250) — compile-verified
//
#include <hip/hip_runtime.h>
#include <math.h>

// fp32 WMMA path on gfx1250: workload is HBM-bound (~205 GFLOP vs ~350MB of
// traffic -> ~15us floor at 23.3 TB/s), so keep reference fp32 numerics and
// use V_WMMA_F32_16X16X4_F32 everywhere. LDS staging goes through the CDNA5
// async global->LDS engine (ASYNCcnt) when the builtin is available.

typedef float v2f __attribute__((ext_vector_type(2)));
typedef float v8f __attribute__((ext_vector_type(8)));
typedef int   v4i __attribute__((ext_vector_type(4)));

typedef v4i __attribute__((address_space(1))) gv4i;  // global int4
typedef v4i __attribute__((address_space(3))) sv4i;  // LDS int4

#define WMMA_F32(a, b, c) \
  __builtin_amdgcn_wmma_f32_16x16x4_f32(false, (a), false, (b), (short)0, (c), false, false)

#if defined(__has_builtin)
#if __has_builtin(__builtin_amdgcn_global_load_async_to_lds_b128)
#define HAVE_ASYNC_LDS 1
#endif
#endif
#ifndef HAVE_ASYNC_LDS
#define HAVE_ASYNC_LDS 0
#endif

// 16-byte global -> LDS copy, async engine if available.
__device__ __forceinline__ void cp16_g2s(const float* g, float* l) {
#if HAVE_ASYNC_LDS
  __builtin_amdgcn_global_load_async_to_lds_b128((gv4i*)g, (sv4i*)l, 0, 0);
#else
  *(float4*)l = *(const float4*)g;
#endif
}

__device__ __forceinline__ void cp_wait() {
#if HAVE_ASYNC_LDS
#if __has_builtin(__builtin_amdgcn_s_wait_asynccnt)
  __builtin_amdgcn_s_wait_asynccnt(0);
#else
  asm volatile("s_wait_asynccnt 0" ::: "memory");
#endif
#endif
}

constexpr int B_ = 2, S_ = 2048, HID_ = 2048, NH_ = 16, HD_ = 128;
constexpr int RING_ = 4, CH_ = 512, SEG_ = 2048;  // DIL = 2 (parity mask)

// ---------------------------------------------------------------------------
// Generic fp32 GEMM: C[M,N] = A[M,K] @ B[K,N]. Block tile 64x128, K-slab 32.
// 256 threads = 8 wave32; each wave owns a 32x32 tile (2x2 WMMA 16x16 blocks).
// ---------------------------------------------------------------------------
constexpr int AS_ = 36;   // A LDS row stride (floats): distinct banks, 16B aligned
constexpr int BS_ = 132;  // B LDS row stride

__global__ __launch_bounds__(256)
void gemm_f32_wmma(const float* __restrict__ A, const float* __restrict__ Bm,
                   float* __restrict__ C, int M, int N, int K) {
  __shared__ float Asl[64 * AS_];
  __shared__ float Bsl[32 * BS_];

  const int tid  = threadIdx.x;
  const int lane = tid & 31;
  const int wv   = tid >> 5;
  const int lr   = lane & 15;
  const int hf   = lane >> 4;

  const int bM = (blockIdx.x % (M / 64)) * 64;
  const int bN = (blockIdx.x / (M / 64)) * 128;
  const int wM = (wv >> 2) * 32;
  const int wN = (wv & 3) * 32;

  v8f acc[2][2];
#pragma unroll
  for (int m = 0; m < 2; ++m)
#pragma unroll
    for (int n = 0; n < 2; ++n)
      acc[m][n] = (v8f){0, 0, 0, 0, 0, 0, 0, 0};

  for (int k0 = 0; k0 < K; k0 += 32) {
    // stage A slab: 64 x 32 (async global->LDS, 16B per lane)
#pragma unroll
    for (int j = 0; j < 2; ++j) {
      int idx = tid * 2 + j;                 // 0..511 float4s
      int r = idx >> 3, c4 = (idx & 7) * 4;
      cp16_g2s(A + (size_t)(bM + r) * K + k0 + c4, &Asl[r * AS_ + c4]);
    }
    // stage B slab: 32 x 128
#pragma unroll
    for (int j = 0; j < 4; ++j) {
      int idx = tid * 4 + j;                 // 0..1023 float4s
      int r = idx >> 5, c4 = (idx & 31) * 4;
      cp16_g2s(Bm + (size_t)(k0 + r) * N + bN + c4, &Bsl[r * BS_ + c4]);
    }
    // prefetch next K slab into cache (global_prefetch_b8)
    if (k0 + 32 < K) {
      __builtin_prefetch(A + (size_t)(bM + (tid >> 2)) * K + k0 + 32, 0, 1);
      __builtin_prefetch(Bm + (size_t)(k0 + 32 + (tid >> 3)) * N + bN + (tid & 7) * 16, 0, 1);
    }
    cp_wait();
    __syncthreads();

#pragma unroll
    for (int t = 0; t < 8; ++t) {
      const int kb = t * 4 + hf * 2;         // A/B frag K base for this half
      v2f af[2], bf[2];
#pragma unroll
      for (int m = 0; m < 2; ++m) {
        const int row = wM + m * 16 + lr;
        af[m] = *(const v2f*)(&Asl[row * AS_ + kb]);
      }
#pragma unroll
      for (int n = 0; n < 2; ++n) {
        const int col = wN + n * 16 + lr;
        v2f t2;
        t2[0] = Bsl[kb * BS_ + col];
        t2[1] = Bsl[(kb + 1) * BS_ + col];
        bf[n] = t2;
      }
#pragma unroll
      for (int m = 0; m < 2; ++m)
#pragma unroll
        for (int n = 0; n < 2; ++n)
          acc[m][n] = WMMA_F32(af[m], bf[n], acc[m][n]);
    }
    __syncthreads();
  }

#pragma unroll
  for (int m = 0; m < 2; ++m)
#pragma unroll
    for (int n = 0; n < 2; ++n)
#pragma unroll
      for (int i = 0; i < 8; ++i) {
        int row = bM + wM + m * 16 + i + 8 * hf;   // C layout: M = i + 8*half
        int col = bN + wN + n * 16 + lr;           // N = lane%16
        C[(size_t)row * N + col] = acc[m][n][i];
      }
}

// ---------------------------------------------------------------------------
// Ring attention with per-kv-chunk (non-rescaled) softmax.
// One WG = (b, h, 64 q-rows); 4 wave32, each wave owns 16 q-rows.
// Pass A per chunk: WMMA scores -> masked row max (shfl reduce, no storage).
// Pass B per chunk: recompute scores (compute is free, LDS is not), bounce
// each 16x16 block via tiny LDS buffer to A-frag layout, exp, accumulate
// den and num += e @ V with 8 WMMA accumulators (16 x 128).
// ---------------------------------------------------------------------------
__global__ __launch_bounds__(128)
void attn_ring(const float* __restrict__ qkv, float* __restrict__ out) {
  __shared__ float Qs[64 * 132];        // raw Q tile (scores scaled post-WMMA)
  __shared__ float Eb[4][16 * 20];      // per-wave 16x16 score block, padded
  __shared__ float rmx[64];             // per-row chunk max
  __shared__ float dns[64];             // per-row denominator

  const int tid  = threadIdx.x;
  const int lane = tid & 31;
  const int wv   = tid >> 5;
  const int lr   = lane & 15;
  const int hf   = lane >> 4;

  const int nQT = S_ / 64;
  const int bh  = blockIdx.x / nQT;
  const int qt  = blockIdx.x % nQT;
  const int b   = bh / NH_;
  const int h   = bh % NH_;
  const int q0  = qt * 64;

  const size_t rowstride = 3 * (size_t)HID_;  // qkv row pitch
  const float* qbase = qkv + (size_t)b * S_ * rowstride + h * HD_;
  const float* kbase = qbase + HID_;
  const float* vbase = qbase + 2 * HID_;

  // stage Q tile (64 x 128) into LDS via the async engine
#pragma unroll
  for (int j = 0; j < 16; ++j) {
    int idx = tid + j * 128;                  // 0..2047 float4s
    int r = idx >> 5, c4 = (idx & 31) * 4;
    cp16_g2s(qbase + (size_t)(q0 + r) * rowstride + c4, &Qs[r * 132 + c4]);
  }
  cp_wait();
  __syncthreads();

  const float rscale = rsqrtf((float)HD_);
  float denAcc = 0.f;
  v8f num[8];
#pragma unroll
  for (int nb = 0; nb < 8; ++nb) num[nb] = (v8f){0, 0, 0, 0, 0, 0, 0, 0};

  const int qw = q0 + wv * 16;   // wave's first global q row

  for (int rc = 0; rc < RING_; ++rc) {
    const int kv0 = rc * CH_;

    // ---- Pass A: per-chunk masked row max -------------------------------
    float rmax8[8];
#pragma unroll
    for (int i = 0; i < 8; ++i) rmax8[i] = -3.0e38f;

    for (int c0 = 0; c0 < CH_; c0 += 16) {
      v8f sc = (v8f){0, 0, 0, 0, 0, 0, 0, 0};
      const float* kp = kbase + (size_t)(kv0 + c0 + lr) * rowstride;
#pragma unroll
      for (int t = 0; t < HD_ / 4; ++t) {
        const int kb = t * 4 + hf * 2;
        v2f af = *(const v2f*)(&Qs[(wv * 16 + lr) * 132 + kb]);
        v2f bf = *(const v2f*)(kp + kb);
        sc = WMMA_F32(af, bf, sc);
      }
      const int kcol = kv0 + c0 + lr;
#pragma unroll
      for (int i = 0; i < 8; ++i) {
        const int qrow = qw + i + 8 * hf;
        const bool ok = ((qrow / SEG_) == (kcol / SEG_)) && (((qrow ^ kcol) & 1) == 0);
        float s = ok ? sc[i] * rscale : -3.0e38f;
        s = fmaxf(s, __shfl_xor(s, 1, 32));
        s = fmaxf(s, __shfl_xor(s, 2, 32));
        s = fmaxf(s, __shfl_xor(s, 4, 32));
        s = fmaxf(s, __shfl_xor(s, 8, 32));   // max over the 16 lanes of this half
        rmax8[i] = fmaxf(rmax8[i], s);
      }
    }
#pragma unroll
    for (int i = 0; i < 8; ++i)
      if (lr == i) rmx[wv * 16 + 8 * hf + i] = rmax8[i];
    __syncthreads();

    // ---- Pass B: recompute scores, exp, accumulate num/den ---------------
    for (int c0 = 0; c0 < CH_; c0 += 16) {
      v8f sc = (v8f){0, 0, 0, 0, 0, 0, 0, 0};
      const float* kp = kbase + (size_t)(kv0 + c0 + lr) * rowstride;
#pragma unroll
      for (int t = 0; t < HD_ / 4; ++t) {
        const int kb = t * 4 + hf * 2;
        v2f af = *(const v2f*)(&Qs[(wv * 16 + lr) * 132 + kb]);
        v2f bf = *(const v2f*)(kp + kb);
        sc = WMMA_F32(af, bf, sc);
      }
      const int kcol = kv0 + c0 + lr;
#pragma unroll
      for (int i = 0; i < 8; ++i) {
        const int qrow = qw + i + 8 * hf;
        const bool ok = ((qrow / SEG_) == (kcol / SEG_)) && (((qrow ^ kcol) & 1) == 0);
        Eb[wv][(i + 8 * hf) * 20 + lr] = ok ? sc[i] * rscale : -3.0e38f;
      }
      __syncthreads();   // C-layout -> A-frag layout bounce visible

      const float mrow = rmx[wv * 16 + lr];
#pragma unroll
      for (int kk = 0; kk < 4; ++kk) {
        const int kb = kk * 4 + hf * 2;
        const float e0 = __expf(Eb[wv][lr * 20 + kb]     - mrow);
        const float e1 = __expf(Eb[wv][lr * 20 + kb + 1] - mrow);
        denAcc += e0 + e1;                     // each (row,col) counted once
        v2f af; af[0] = e0; af[1] = e1;
        const float* vp0 = vbase + (size_t)(kv0 + c0 + kb) * rowstride;
        const float* vp1 = vp0 + rowstride;
#pragma unroll
        for (int nb = 0; nb < 8; ++nb) {
          const int d = nb * 16 + lr;
          v2f bf; bf[0] = vp0[d]; bf[1] = vp1[d];
          num[nb] = WMMA_F32(af, bf, num[nb]);
        }
      }
      __syncthreads();   // Eb reuse across c0 iterations
    }
  }

  // den per row: lane l covers row lr; combine both halves
  denAcc += __shfl_xor(denAcc, 16, 32);
  if (hf == 0) dns[wv * 16 + lr] = denAcc;
  __syncthreads();

#pragma unroll
  for (int i = 0; i < 8; ++i) {
    const int qrow = qw + i + 8 * hf;
    const float inv = 1.0f / (dns[wv * 16 + i + 8 * hf] + 1e-8f);
#pragma unroll
    for (int nb = 0; nb < 8; ++nb) {
      const int d = h * HD_ + nb * 16 + lr;
      out[(size_t)(b * S_ + qrow) * HID_ + d] = num[nb][i] * inv;
    }
  }
}

// ---------------------------------------------------------------------------
extern "C" void kernel_launch(void* const* d_in, const int* in_sizes, int n_in,
                              void* d_out, int out_size, void* d_ws, size_t ws_size,
                              hipStream_t stream) {
  const float* x    = (const float*)d_in[0];  // [B*S, HID]
  const float* Wqkv = (const float*)d_in[1];  // [HID, 3*HID]
  const float* Wout = (const float*)d_in[2];  // [HID, HID]
  float* out = (float*)d_out;                 // [B*S, HID]

  float* qkv = (float*)d_ws;                                   // [B*S, 3*HID]
  float* att = qkv + (size_t)B_ * S_ * 3 * HID_;               // [B*S, HID]

  // 1) qkv = x @ W_qkv
  {
    const int M = B_ * S_, K = HID_, N = 3 * HID_;
    gemm_f32_wmma<<<dim3((M / 64) * (N / 128)), 256, 0, stream>>>(x, Wqkv, qkv, M, N, K);
  }
  // 2) ring attention with per-chunk softmax
  attn_ring<<<dim3(B_ * NH_ * (S_ / 64)), 128, 0, stream>>>(qkv, att);
  // 3) out = att @ W_out
  {
    const int M = B_ * S_, K = HID_, N = HID_;
    gemm_f32_wmma<<<dim3((M / 64) * (N / 128)), 256, 0, stream>>>(att, Wout, out, M, N, K);
  }
}